// PCA_57432302682837
// MI455X (gfx1250) — compile-verified
//
#include <hip/hip_runtime.h>
#include <math.h>

typedef __attribute__((ext_vector_type(2))) float v2f;
typedef __attribute__((ext_vector_type(8))) float v8f;

#define GROUPS_PER_BLOCK 32
#define LDS_STRIDE 196                      // 192 + 4 pad; 196 % 64 == 4 -> conflict-free group stride
#define ONES_BASE (GROUPS_PER_BLOCK * LDS_STRIDE)   // 192-float region of 1.0f
#define GRAM_BASE (ONES_BASE + 192)
#define LDS_FLOATS (GRAM_BASE + GROUPS_PER_BLOCK * 16)

__device__ __forceinline__ float3 mk3(float x, float y, float z) {
  float3 r; r.x = x; r.y = y; r.z = z; return r;
}
__device__ __forceinline__ float3 crossv(float3 a, float3 b) {
  return mk3(a.y * b.z - a.z * b.y, a.z * b.x - a.x * b.z, a.x * b.y - a.y * b.x);
}
__device__ __forceinline__ float dotv(float3 a, float3 b) {
  return a.x * b.x + a.y * b.y + a.z * b.z;
}
// Eigenvector of symmetric G for eigenvalue lam: largest cross product of rows of (G - lam*I)
__device__ __forceinline__ float3 eigvec3(float gxx, float gxy, float gxz,
                                          float gyy, float gyz, float gzz, float lam) {
  float3 r0 = mk3(gxx - lam, gxy, gxz);
  float3 r1 = mk3(gxy, gyy - lam, gyz);
  float3 r2 = mk3(gxz, gyz, gzz - lam);
  float3 c0 = crossv(r0, r1), c1 = crossv(r0, r2), c2 = crossv(r1, r2);
  float n0 = dotv(c0, c0), n1 = dotv(c1, c1), n2 = dotv(c2, c2);
  float3 best = c0; float nb = n0;
  if (n1 > nb) { best = c1; nb = n1; }
  if (n2 > nb) { best = c2; nb = n2; }
  if (nb < 1e-30f) return mk3(1.f, 0.f, 0.f);
  float inv = rsqrtf(nb);
  return mk3(best.x * inv, best.y * inv, best.z * inv);
}

__global__ __launch_bounds__(256) void pca_feat_kernel(
    const float* __restrict__ group_xyz,  // (BN, 64, 3) contiguous
    const float* __restrict__ new_xyz,    // (BN, 3)
    float* __restrict__ out,              // (BN, 15)
    int num_groups) {
  __shared__ float lds[LDS_FLOATS];  // data | ones | gram scratch  (~27.9 KB)

  const int tid = threadIdx.x;
  const int lane = tid & 31;
  const int wave = tid >> 5;
  const int blockGroupBase = blockIdx.x * GROUPS_PER_BLOCK;

  // ---- Stage 1: coalesced b128 global -> padded LDS; fill ones region ----
  if (tid < 192) lds[ONES_BASE + tid] = 1.0f;
  {
    const size_t baseF = (size_t)blockGroupBase * 192;
    const size_t totalF4 = ((size_t)num_groups * 192) >> 2;
    const float4* src = (const float4*)group_xyz + (baseF >> 2);
#pragma unroll
    for (int i = 0; i < 6; ++i) {
      int idx = tid + i * 256;  // 0..1535 float4s = 32 groups
      if ((baseF >> 2) + (size_t)idx < totalF4) {
        float4 v = src[idx];
        int fi = idx << 2;
        int g = fi / 192;             // group in block (float4 never crosses: 192%4==0)
        int off = fi - g * 192;
        *(float4*)&lds[g * LDS_STRIDE + off] = v;  // (g*196+off)*4 is 16B aligned
      }
    }
  }
  __syncthreads();

  // ---- Stage 2: augmented Gram via V_WMMA_F32_16X16X4_F32, 4 groups per wave ----
  // Column map j in [0,16): group gl=j>>2, coord c=j&3; c==3 is the "ones" column so the
  // 4x4 diagonal blocks of D = Xa^T * Xa hold [Gram(3x3) | Sx Sy Sz | 64].
  // A-fragment (16x4 f32, 2 VGPRs): lane<16 -> k0,k1 ; lane>=16 -> k2,k3. B == A here.
  // Ones lanes read a shared LDS region of 1.0f -> all ds_loads unconditional (no EXEC masking),
  // pairs (12s, 12s+3 dwords) fuse into ds_load_2addr_b32.
  const int j = lane & 15;
  const int hi = lane >> 4;
  const int gl = j >> 2;
  const int c = j & 3;
  const int gbase = (c == 3) ? (ONES_BASE + hi * 6)
                             : ((wave * 4 + gl) * LDS_STRIDE + hi * 6 + c);

  float f0[16], f1[16];
#pragma unroll
  for (int s = 0; s < 16; ++s) {
    f0[s] = lds[gbase + s * 12];      // k = 4s+2hi
    f1[s] = lds[gbase + s * 12 + 3];  // k = 4s+2hi+1
  }

  v8f acc = {0.f, 0.f, 0.f, 0.f, 0.f, 0.f, 0.f, 0.f};
#pragma unroll
  for (int s = 0; s < 16; ++s) {  // K = 64 = 16 steps * 4
    v2f a = {f0[s], f1[s]};
    acc = __builtin_amdgcn_wmma_f32_16x16x4_f32(
        /*neg_a=*/false, a, /*neg_b=*/false, a,
        /*c_mod=*/(short)0, acc, /*reuse_a=*/false, /*reuse_b=*/false);
  }

  // Harvest diagonal 4x4 blocks: lane holds column n=j (rows 0..7 lo-half, 8..15 hi-half)
  if ((gl >> 1) == hi) {
    float r0, r1, r2, r3;
    if ((gl & 1) == 0) { r0 = acc.s0; r1 = acc.s1; r2 = acc.s2; r3 = acc.s3; }
    else               { r0 = acc.s4; r1 = acc.s5; r2 = acc.s6; r3 = acc.s7; }
    float* sc = &lds[GRAM_BASE + (wave * 4 + gl) * 16];
    sc[0 + c] = r0; sc[4 + c] = r1; sc[8 + c] = r2; sc[12 + c] = r3;
  }
  __syncthreads();

  // ---- Stage 3: 3x3 eigensolve + features, lanes 0..3 of each wave ----
  if (lane < 4) {
    const int glocal = wave * 4 + lane;
    const int grp = blockGroupBase + glocal;
    if (grp < num_groups) {
      const float* sc = &lds[GRAM_BASE + glocal * 16];
      float gxx = sc[0], gxy = sc[1], gxz = sc[2],  Sx = sc[3];
      float gyy = sc[5], gyz = sc[6],  Sy = sc[7];
      float gzz = sc[10], Sz = sc[11];

      // Smith's analytic eigenvalues of symmetric 3x3 (descending l1>=l2>=l3)
      float q = (gxx + gyy + gzz) * (1.f / 3.f);
      float p1 = gxy * gxy + gxz * gxz + gyz * gyz;
      float d0 = gxx - q, d1 = gyy - q, d2 = gzz - q;
      float p2 = d0 * d0 + d1 * d1 + d2 * d2 + 2.f * p1;
      float l1, l2, l3;
      if (p2 < 1e-22f) {
        l1 = l2 = l3 = q;
      } else {
        float p = sqrtf(p2 * (1.f / 6.f));
        float ip = 1.f / p;
        float b00 = d0 * ip, b11 = d1 * ip, b22 = d2 * ip;
        float b01 = gxy * ip, b02 = gxz * ip, b12 = gyz * ip;
        float detB = b00 * (b11 * b22 - b12 * b12)
                   - b01 * (b01 * b22 - b12 * b02)
                   + b02 * (b01 * b12 - b11 * b02);
        float r = fminf(1.f, fmaxf(-1.f, 0.5f * detB));
        float phi = acosf(r) * (1.f / 3.f);
        l1 = q + 2.f * p * cosf(phi);
        l3 = q + 2.f * p * cosf(phi + 2.0943951023931953f);  // +2*pi/3
        l2 = 3.f * q - l1 - l3;
      }
      float a1 = sqrtf(fmaxf(l1, 0.f));
      float a2 = sqrtf(fmaxf(l2, 0.f));
      float a3 = sqrtf(fmaxf(l3, 0.f));

      float3 v1 = eigvec3(gxx, gxy, gxz, gyy, gyz, gzz, l1);
      float3 v2 = eigvec3(gxx, gxy, gxz, gyy, gyz, gzz, l2);
      // Gram-Schmidt v2 against v1, with fallback to any perpendicular
      float dp = dotv(v1, v2);
      v2 = mk3(v2.x - dp * v1.x, v2.y - dp * v1.y, v2.z - dp * v1.z);
      float n2 = dotv(v2, v2);
      if (n2 < 1e-12f) {
        float3 ax = (fabsf(v1.x) < 0.9f) ? mk3(1.f, 0.f, 0.f) : mk3(0.f, 1.f, 0.f);
        v2 = crossv(v1, ax);
        n2 = dotv(v2, v2);
      }
      float inv2 = rsqrtf(fmaxf(n2, 1e-30f));
      v2 = mk3(v2.x * inv2, v2.y * inv2, v2.z * inv2);
      float3 v3 = crossv(v1, v2);

      float inv1 = 1.f / (a1 + 1e-10f);
      float* o = out + (size_t)grp * 15;
      o[0] = (a1 - a2) * inv1;                                        // linearity
      o[1] = (a2 - a3) * inv1;                                        // planarity
      o[2] = a3 * inv1;                                               // scattering
      o[3] = a1 * fabsf(v1.x) + a2 * fabsf(v2.x) + a3 * fabsf(v3.x);  // direction
      o[4] = a1 * fabsf(v1.y) + a2 * fabsf(v2.y) + a3 * fabsf(v3.y);
      o[5] = a1 * fabsf(v1.z) + a2 * fabsf(v2.z) + a3 * fabsf(v3.z);
      o[6] = new_xyz[3 * (size_t)grp + 0];                            // mean_xyz
      o[7] = new_xyz[3 * (size_t)grp + 1];
      o[8] = new_xyz[3 * (size_t)grp + 2];
      const float iK = 1.f / 64.f, iKm1 = 1.f / 63.f;                 // std, ddof=1
      o[9]  = sqrtf(fmaxf(0.f, (gxx - Sx * Sx * iK) * iKm1));
      o[10] = sqrtf(fmaxf(0.f, (gyy - Sy * Sy * iK) * iKm1));
      o[11] = sqrtf(fmaxf(0.f, (gzz - Sz * Sz * iK) * iKm1));
      o[12] = v1.x; o[13] = v1.y; o[14] = v1.z;                       // norm = vh[:,0]
    }
  }
}

extern "C" void kernel_launch(void* const* d_in, const int* in_sizes, int n_in,
                              void* d_out, int out_size, void* d_ws, size_t ws_size,
                              hipStream_t stream) {
  const float* group_xyz = (const float*)d_in[0];
  const float* new_xyz = (const float*)d_in[1];
  float* out = (float*)d_out;
  const int num_groups = in_sizes[1] / 3;  // B*N
  const int blocks = (num_groups + GROUPS_PER_BLOCK - 1) / GROUPS_PER_BLOCK;
  pca_feat_kernel<<<blocks, 256, 0, stream>>>(group_xyz, new_xyz, out, num_groups);
}